// RelativeInformationInjection_31817117729123
// MI455X (gfx1250) — compile-verified
//
#include <hip/hip_runtime.h>

// Problem constants (fixed by the reference).
#define B_    4
#define SEQ_  4096
#define D_    64
#define BS_   32
#define V_    512
#define R_    128         // SEQ / BS
#define TRI_  8256        // R*(R+1)/2  (tril size per batch)
#define CHUNK_ 8          // p's served per gather block

typedef __attribute__((ext_vector_type(2))) float v2f;
typedef __attribute__((ext_vector_type(8))) float v8f;

// ---------------------------------------------------------------------------
// Kernel 1: scores[b,m,v] = sum_d q[b,m,d] * emb[b,v,d]   (fp32 WMMA)
// One wave computes a 16(M) x 16(N) tile; K-loop of 16 x V_WMMA_F32_16X16X4_F32.
// Block = 128 threads = 4 waves, each wave takes one N-tile (A rows shared in L0).
// ---------------------------------------------------------------------------
__global__ __launch_bounds__(128)
void scores_wmma(const float* __restrict__ q,
                 const float* __restrict__ emb,
                 float* __restrict__ scores) {
  const int lane = threadIdx.x & 31;
  const int wv   = threadIdx.x >> 5;            // 0..3
  const int mt   = blockIdx.x;                  // 0..SEQ/16-1
  const int nt   = blockIdx.y * 4 + wv;         // 0..V/16-1
  const int b    = blockIdx.z;
  const int half = lane >> 4;                   // K-half selector
  const int idx  = lane & 15;                   // row (A) / col (B)

  // A lane view: row (mt*16+idx), cols k + 2*half + {0,1}
  const float* qrow = q   + ((size_t)b * SEQ_ + (size_t)mt * 16 + idx) * D_ + 2 * half;
  // B lane view: B[k][n] = emb[b, nt*16+n, k] -> same addressing pattern
  const float* erow = emb + ((size_t)b * V_   + (size_t)nt * 16 + idx) * D_ + 2 * half;

  v8f acc = {};
#pragma unroll
  for (int k = 0; k < D_; k += 4) {
    v2f a, bb;
    a.x  = qrow[k + 0];  a.y  = qrow[k + 1];
    bb.x = erow[k + 0];  bb.y = erow[k + 1];
    // 8 args: (neg_a, A, neg_b, B, c_mod, C, reuse_a, reuse_b)
    acc = __builtin_amdgcn_wmma_f32_16x16x4_f32(false, a, false, bb,
                                                (short)0, acc, false, false);
  }

  // C/D layout: VGPR i -> (M = 8*half + i, N = idx)
  float* out = scores + ((size_t)b * SEQ_ + (size_t)mt * 16 + 8 * half) * V_
                      + (size_t)nt * 16 + idx;
#pragma unroll
  for (int i = 0; i < 8; ++i)
    out[(size_t)i * V_] = acc[i];
}

// ---------------------------------------------------------------------------
// Kernel 2: grouped gather. tril structure => p = b*TRI_ + r*(r+1)/2 + c,
// row r has (r+1) occurrences. One block stages the 32x512 score slice for
// (b, r) into LDS (64 KB) and serves a chunk of up to CHUNK_ p's:
//   out[p,i,j] = slice[i, info[p,i,j]]
// info reads int4-coalesced, out writes float4-coalesced, gathers hit LDS.
// ---------------------------------------------------------------------------
__global__ __launch_bounds__(256)
void gather_inject(const float* __restrict__ scores,
                   const int* __restrict__ info,
                   float* __restrict__ out) {
  const int b   = blockIdx.x >> 7;        // 0..3
  const int r   = blockIdx.x & 127;       // 0..127
  const int cnt = r + 1;                  // occurrences of this (b, r)
  const int c0  = blockIdx.y * CHUNK_;
  if (c0 >= cnt) return;

  __shared__ float tile[BS_ * V_];        // 32 x 512 f32 = 64 KB

  // Stage the score slice: 4096 float4's with 256 threads (16 iters).
  const float4* src =
      (const float4*)(scores + ((size_t)b * SEQ_ + (size_t)r * BS_) * V_);
  for (int t = threadIdx.x; t < (BS_ * V_) / 4; t += 256)
    ((float4*)tile)[t] = src[t];
  __syncthreads();

  const int    cend  = (c0 + CHUNK_ < cnt) ? (c0 + CHUNK_) : cnt;
  const size_t pbase = (size_t)b * TRI_ + ((size_t)r * (r + 1)) / 2;
  const int    t     = threadIdx.x;       // each thread: 4 consecutive j's
  const int    row   = t >> 3;            // i = (4*t) / 32

  for (int c = c0; c < cend; ++c) {
    const size_t p   = pbase + c;
    const size_t off = p * (size_t)(BS_ * BS_);
    int4 iv = ((const int4*)(info + off))[t];
    float4 ov;
    ov.x = tile[row * V_ + iv.x];
    ov.y = tile[row * V_ + iv.y];
    ov.z = tile[row * V_ + iv.z];
    ov.w = tile[row * V_ + iv.w];
    ((float4*)(out + off))[t] = ov;
  }
}

// ---------------------------------------------------------------------------
extern "C" void kernel_launch(void* const* d_in, const int* in_sizes, int n_in,
                              void* d_out, int out_size, void* d_ws, size_t ws_size,
                              hipStream_t stream) {
  (void)in_sizes; (void)n_in; (void)out_size; (void)ws_size;
  const float* q    = (const float*)d_in[0];
  const float* emb  = (const float*)d_in[1];
  const int*   info = (const int*)d_in[2];
  // d_in[3]=idxs_batch, d_in[4]=idxs_row: implied by the closed-form tril layout.
  float* out    = (float*)d_out;
  float* scores = (float*)d_ws;           // needs B*SEQ*V*4 = 32 MB scratch

  dim3 g1(SEQ_ / 16, V_ / (16 * 4), B_);  // (256, 8, 4), 4 waves/block
  scores_wmma<<<g1, 128, 0, stream>>>(q, emb, scores);

  dim3 g2(B_ * R_, (R_ + CHUNK_ - 1) / CHUNK_, 1);  // (512, 16)
  gather_inject<<<g2, 256, 0, stream>>>(scores, info, out);
}